// DistanceWeightedSelfAttention_84327387890255
// MI455X (gfx1250) — compile-verified
//
#include <hip/hip_runtime.h>

typedef float v2f __attribute__((ext_vector_type(2)));
typedef float v8f __attribute__((ext_vector_type(8)));

#define BATCH 8
#define SEQ 2048
#define DMODEL 256
#define THREADS 128
#define WAVES 4             // THREADS / 32
#define ROWS_PER_WAVE 16
#define ROWS_PER_BLOCK 64   // WAVES * ROWS_PER_WAVE
#define LAMBDA 0.5f
#define NEG_BIG -1.0e9f

__global__ __launch_bounds__(THREADS)
void dwsa_kernel(const float* __restrict__ in,   // (B, S, 2) = (size, height)
                 const float* __restrict__ Wq,   // (1, 256)
                 const float* __restrict__ Wk,   // (1, 256)
                 const float* __restrict__ Wv,   // (1, 256)
                 float* __restrict__ out)        // (B, S, 256)
{
    __shared__ __align__(16) float2 s_kv[SEQ];          // (z, h) per position: 16 KB
    __shared__ float s_red[WAVES][2][ROWS_PER_WAVE];    // [wave][eh|e][row]
    __shared__ float s_A[ROWS_PER_BLOCK];

    const int tid  = threadIdx.x;
    const int b    = blockIdx.x >> 5;                 // 32 blocks per batch
    const int rowBase = (blockIdx.x & 31) * ROWS_PER_BLOCK;
    const int wave = tid >> 5;
    const int lane = tid & 31;
    const int half = lane >> 4;                       // lane half (WMMA K split)
    const int r    = lane & 15;                       // row within 16-row tile
    const int n    = lane & 15;                       // D-matrix column index

    // Cooperative load of this batch's sequence (z,h) into LDS (b64 DS stores).
    const float2* seq = (const float2*)in + (size_t)b * SEQ;
    for (int t = tid; t < SEQ; t += THREADS) s_kv[t] = seq[t];

    // Scalar coupling c = dot(Wq, Wk) / sqrt(D). Redundant per thread; L2-hot.
    float c = 0.0f;
    #pragma unroll 8
    for (int i = 0; i < DMODEL; ++i) c += Wq[i] * Wk[i];
    c *= (1.0f / 16.0f);

    __syncthreads();

    const int   row = rowBase + wave * ROWS_PER_WAVE + r;
    const float2 q2 = s_kv[row];
    const float zs = q2.x, hs = q2.y;

    // ---- Pass 1: row max over all keys. Halves scan disjoint key ranges,
    // combined with a wave32 xor-shuffle so both halves share the row max.
    float m = -3.4e38f;
    const int t0 = half * (SEQ / 2);
    for (int t = t0; t < t0 + SEQ / 2; ++t) {
        float2 kt = s_kv[t];
        float l = c * hs * kt.y - LAMBDA * fabsf(zs - kt.x);
        l += (kt.y == 0.0f) ? NEG_BIG : 0.0f;   // key mask
        m = fmaxf(m, l);
    }
    m = fmaxf(m, __shfl_xor(m, 16, 32));

    // ---- Pass 2: WMMA f32 16x16x4 accumulation of [sum(e*h), sum(e)].
    // A (16x4): lane half 0 -> K{0,1}, half 1 -> K{2,3}; values = exp weights.
    // B (4x16): column 0 = h_key, column 1 = 1, rest 0.
    // D (16x16 f32): N=0 column accumulates sum(e*h), N=1 accumulates sum(e).
    v8f acc = {0.f, 0.f, 0.f, 0.f, 0.f, 0.f, 0.f, 0.f};
    const float4* kv4 = (const float4*)s_kv;
    #pragma unroll 2
    for (int kc = 0; kc < SEQ; kc += 4) {
        const int ka = kc + 2 * half;                 // first of this lane's 2 keys
        float4 kk = kv4[ka >> 1];                     // (z_a, h_a, z_b, h_b), 16B aligned
        const float za = kk.x, ha = kk.y, zb = kk.z, hb = kk.w;
        float la = c * hs * ha - LAMBDA * fabsf(zs - za) + ((ha == 0.0f) ? NEG_BIG : 0.0f);
        float lb = c * hs * hb - LAMBDA * fabsf(zs - zb) + ((hb == 0.0f) ? NEG_BIG : 0.0f);
        v2f a;
        a.x = __expf(la - m);
        a.y = __expf(lb - m);
        v2f bop;
        bop.x = (n == 0) ? ha : ((n == 1) ? 1.0f : 0.0f);
        bop.y = (n == 0) ? hb : ((n == 1) ? 1.0f : 0.0f);
        acc = __builtin_amdgcn_wmma_f32_16x16x4_f32(
            /*neg_a=*/false, a, /*neg_b=*/false, bop,
            /*c_mod=*/(short)0, acc, /*reuse_a=*/false, /*reuse_b=*/false);
    }

    // ---- Drain D columns 0 (eh) and 1 (e) to LDS. Lanes 0-15 hold M=v,
    // lanes 16-31 hold M=v+8 (ISA C/D layout).
    if (n < 2) {
        #pragma unroll
        for (int v = 0; v < 8; ++v)
            s_red[wave][n][v + 8 * half] = acc[v];
    }
    __syncthreads();

    if (tid < ROWS_PER_BLOCK) {
        const int w = tid >> 4, rr = tid & 15;
        const float eh = s_red[w][0][rr];
        const float se = s_red[w][1][rr];   // >= 1 always (max element contributes exp(0))
        s_A[tid] = eh / se;
    }
    __syncthreads();

    // ---- Epilogue: out[b, row, :] = A * Wv. 128 threads x float2 = one 1KB row,
    // fully coalesced; nontemporal stores keep the 16MB output out of L2.
    const float wv0 = Wv[2 * tid];
    const float wv1 = Wv[2 * tid + 1];
    v2f* o2 = (v2f*)out + (size_t)(b * SEQ + rowBase) * (DMODEL / 2) + tid;
    for (int rr = 0; rr < ROWS_PER_BLOCK; ++rr) {
        const float A = s_A[rr];
        v2f val;
        val.x = A * wv0;
        val.y = A * wv1;
        __builtin_nontemporal_store(val, o2);
        o2 += DMODEL / 2;
    }
}

extern "C" void kernel_launch(void* const* d_in, const int* in_sizes, int n_in,
                              void* d_out, int out_size, void* d_ws, size_t ws_size,
                              hipStream_t stream) {
    (void)in_sizes; (void)n_in; (void)out_size; (void)d_ws; (void)ws_size;
    const float* in = (const float*)d_in[0];   // (8, 2048, 2) f32
    const float* Wq = (const float*)d_in[1];   // (1, 256) f32
    const float* Wk = (const float*)d_in[2];   // (1, 256) f32
    const float* Wv = (const float*)d_in[3];   // (1, 256) f32
    float* out = (float*)d_out;                // (8, 2048, 256) f32

    dim3 grid(BATCH * (SEQ / ROWS_PER_BLOCK)); // 256 workgroups
    dim3 block(THREADS);                       // 4 wave32s
    dwsa_kernel<<<grid, block, 0, stream>>>(in, Wq, Wk, Wv, out);
}